// Graph_Pooling_51024211476648
// MI455X (gfx1250) — compile-verified
//
#include <hip/hip_runtime.h>
#include <cstdint>

#ifndef __has_builtin
#define __has_builtin(x) 0
#endif

#define GB 16
#define GC 128
#define GN 4096
#define GK 16
#define NT 256      // n-tile per block (== THREADS)
#define CT 32       // channels per block
#define THREADS 256

typedef int v4i __attribute__((ext_vector_type(4)));
typedef __attribute__((address_space(1))) v4i g_v4i;   // global  int4
typedef __attribute__((address_space(3))) v4i l_v4i;   // LDS     int4

__global__ __launch_bounds__(THREADS, 2) void gpool_kernel(
    const float* __restrict__ x, const int* __restrict__ idx,
    float* __restrict__ out)
{
    __shared__ int s_idx[NT * GK];   // 16 KB of int32 neighbor indices

    const int bid    = blockIdx.x;
    const int cchunk = bid & (GC / CT - 1);          // bits [1:0], 4 chunks
    const int ntile  = (bid >> 2) & (GN / NT - 1);   // bits [5:2], 16 tiles
    const int b      = bid >> 6;                     // bits [9:6], 16 batches

    // ---- Stage index tile (idx[b, ntile*NT .. +NT, :]) into LDS ----
    const int* idx_tile = idx + ((size_t)b * GN + (size_t)ntile * NT) * GK;

#if __has_builtin(__builtin_amdgcn_global_load_async_to_lds_b128) && \
    __has_builtin(__builtin_amdgcn_s_wait_asynccnt)
    {
        g_v4i* gbase = (g_v4i*)idx_tile;   // 16B chunks
        l_v4i* lbase = (l_v4i*)s_idx;
        // NT*GK*4 = 16384 bytes = 1024 chunks; 256 threads x 4 rounds
        #pragma unroll
        for (int r = 0; r < (NT * GK * 4) / (THREADS * 16); ++r) {
            const int chunk = r * THREADS + (int)threadIdx.x;
            __builtin_amdgcn_global_load_async_to_lds_b128(
                gbase + chunk, lbase + chunk, 0, 0);
        }
        __builtin_amdgcn_s_wait_asynccnt(0);   // this wave's copies done
    }
#else
    for (int i = threadIdx.x; i < NT * GK; i += THREADS)
        s_idx[i] = idx_tile[i];
#endif
    __syncthreads();   // all waves' copies visible in LDS

    // ---- Pull this thread's K indices into registers (reused for 32 c's) ----
    const int n_local = threadIdx.x;
    int ind[GK];
    #pragma unroll
    for (int k = 0; k < GK; ++k) ind[k] = s_idx[n_local * GK + k];

    const size_t base = ((size_t)b * GC + (size_t)cchunk * CT) * GN;
    const float* __restrict__ xrow = x + base;
    float* __restrict__ orow       = out + base;
    const int n = ntile * NT + n_local;

    // ---- Main loop: for each channel, 1 coalesced load + 16 gathers + max chain ----
    for (int c = 0; c < CT; ++c) {
        const float* __restrict__ r = xrow + (size_t)c * GN;
        float m = r[n];
        #pragma unroll
        for (int k = 0; k < GK; ++k)
            m = fmaxf(m, r[ind[k]]);
        orow[(size_t)c * GN + n] = m;
    }
}

extern "C" void kernel_launch(void* const* d_in, const int* in_sizes, int n_in,
                              void* d_out, int out_size, void* d_ws, size_t ws_size,
                              hipStream_t stream) {
    (void)in_sizes; (void)n_in; (void)out_size; (void)d_ws; (void)ws_size;
    const float* x   = (const float*)d_in[0];
    const int*   idx = (const int*)d_in[1];
    float*       out = (float*)d_out;

    const int grid = GB * (GN / NT) * (GC / CT);   // 1024 blocks
    gpool_kernel<<<grid, THREADS, 0, stream>>>(x, idx, out);
}